// SNAP_GNN_34840774705776
// MI455X (gfx1250) — compile-verified
//
#include <hip/hip_runtime.h>
#include <hip/hip_bf16.h>

typedef __attribute__((ext_vector_type(2))) float v2f;
typedef __attribute__((ext_vector_type(8))) float v8f;

__device__ __forceinline__ v8f wmma4(v2f a, v2f b, v8f c) {
    // D = A(16x4,f32) * B(4x16,f32) + C(16x16,f32)
    return __builtin_amdgcn_wmma_f32_16x16x4_f32(
        /*neg_a=*/false, a, /*neg_b=*/false, b,
        /*c_mod=*/(short)0, c, /*reuse_a=*/false, /*reuse_b=*/false);
}

// ---------------------------------------------------------------------------
// degree / normalization
// ---------------------------------------------------------------------------
__global__ void k_zero(float* p, int n) {
    int i = blockIdx.x * blockDim.x + threadIdx.x;
    if (i < n) p[i] = 0.0f;
}

__global__ void k_count_deg(const int* __restrict__ dst, float* __restrict__ deg, int E) {
    int e = blockIdx.x * blockDim.x + threadIdx.x;
    if (e < E) atomicAdd(&deg[dst[e]], 1.0f);
}

__global__ void k_dinv(float* __restrict__ d, int N) {
    int i = blockIdx.x * blockDim.x + threadIdx.x;
    if (i < N) d[i] = rsqrtf(d[i] + 1.0f);   // +1 self-loop
}

// ---------------------------------------------------------------------------
// encoder: h0 = relu([x@fcW+fcb , cnn@cnnW+cnnb])   [N,64]
// one wave per 16-node tile; 4 accumulators (64 output cols)
// ---------------------------------------------------------------------------
__global__ void k_encoder(const float* __restrict__ x, const float* __restrict__ cnn,
                          const float* __restrict__ fcW, const float* __restrict__ fcb,
                          const float* __restrict__ cnnW, const float* __restrict__ cnnb,
                          float* __restrict__ h0, int N, int nTiles) {
    int wave = (blockIdx.x * blockDim.x + threadIdx.x) >> 5;
    if (wave >= nTiles) return;
    int lane = threadIdx.x & 31;
    int n    = lane & 15;        // M for A-loads, N for B-loads/stores
    int kh   = lane >> 4;        // K half-select
    int rowA = wave * 16 + n;    if (rowA >= N) rowA = N - 1;

    v8f acc0 = {}, acc1 = {}, acc2 = {}, acc3 = {};

    const float* ax = x + (size_t)rowA * 32;
#pragma unroll
    for (int k0 = 0; k0 < 32; k0 += 4) {
        int kb = k0 + 2 * kh;
        v2f a;  a.x = ax[kb];               a.y = ax[kb + 1];
        v2f b0; b0.x = fcW[kb * 32 + n];    b0.y = fcW[(kb + 1) * 32 + n];
        v2f b1; b1.x = fcW[kb * 32 + 16 + n]; b1.y = fcW[(kb + 1) * 32 + 16 + n];
        acc0 = wmma4(a, b0, acc0);
        acc1 = wmma4(a, b1, acc1);
    }
    const float* ac = cnn + (size_t)rowA * 128;
#pragma unroll
    for (int k0 = 0; k0 < 128; k0 += 4) {
        int kb = k0 + 2 * kh;
        v2f a;  a.x = ac[kb];                a.y = ac[kb + 1];
        v2f b0; b0.x = cnnW[kb * 32 + n];    b0.y = cnnW[(kb + 1) * 32 + n];
        v2f b1; b1.x = cnnW[kb * 32 + 16 + n]; b1.y = cnnW[(kb + 1) * 32 + 16 + n];
        acc2 = wmma4(a, b0, acc2);
        acc3 = wmma4(a, b1, acc3);
    }
#pragma unroll
    for (int v = 0; v < 8; ++v) {
        int mr   = (lane < 16) ? v : v + 8;
        int node = wave * 16 + mr;
        if (node < N) {
            float* o = h0 + (size_t)node * 64;
            o[n]      = fmaxf(acc0[v] + fcb[n],       0.0f);
            o[16 + n] = fmaxf(acc1[v] + fcb[16 + n],  0.0f);
            o[32 + n] = fmaxf(acc2[v] + cnnb[n],      0.0f);
            o[48 + n] = fmaxf(acc3[v] + cnnb[16 + n], 0.0f);
        }
    }
}

// ---------------------------------------------------------------------------
// t1 = h0 @ W1    [N,64] x [64,32] -> [N,32]   (bias folded into agg init)
// ---------------------------------------------------------------------------
__global__ void k_gemm_t1(const float* __restrict__ h0, const float* __restrict__ W1,
                          float* __restrict__ t1, int N, int nTiles) {
    int wave = (blockIdx.x * blockDim.x + threadIdx.x) >> 5;
    if (wave >= nTiles) return;
    int lane = threadIdx.x & 31;
    int n    = lane & 15;
    int kh   = lane >> 4;
    int rowA = wave * 16 + n;    if (rowA >= N) rowA = N - 1;

    v8f acc0 = {}, acc1 = {};
    const float* ar = h0 + (size_t)rowA * 64;
#pragma unroll
    for (int k0 = 0; k0 < 64; k0 += 4) {
        int kb = k0 + 2 * kh;
        v2f a;  a.x = ar[kb];              a.y = ar[kb + 1];
        v2f b0; b0.x = W1[kb * 32 + n];    b0.y = W1[(kb + 1) * 32 + n];
        v2f b1; b1.x = W1[kb * 32 + 16 + n]; b1.y = W1[(kb + 1) * 32 + 16 + n];
        acc0 = wmma4(a, b0, acc0);
        acc1 = wmma4(a, b1, acc1);
    }
#pragma unroll
    for (int v = 0; v < 8; ++v) {
        int mr   = (lane < 16) ? v : v + 8;
        int node = wave * 16 + mr;
        if (node < N) {
            float* o = t1 + (size_t)node * 32;
            o[n]      = acc0[v];
            o[16 + n] = acc1[v];
        }
    }
}

// agg1[i,f] = t1[i,f] * dinv[i]^2 + b1[f]   (self-loop + bias pre-folded)
__global__ void k_init_agg1(const float* __restrict__ t1, const float* __restrict__ dinv,
                            const float* __restrict__ b1, float* __restrict__ agg1, int N) {
    int i = blockIdx.x * blockDim.x + threadIdx.x;
    if (i < N * 32) {
        int node = i >> 5, f = i & 31;
        float di = dinv[node];
        agg1[i] = t1[i] * di * di + b1[f];
    }
}

// one wave per edge, lane = feature (32 features)
__global__ void k_edge1(const int* __restrict__ src, const int* __restrict__ dst,
                        const float* __restrict__ dinv, const float* __restrict__ t1,
                        float* __restrict__ agg1, int E) {
    int e = (blockIdx.x * blockDim.x + threadIdx.x) >> 5;
    if (e >= E) return;
    int lane = threadIdx.x & 31;
    int s = src[e], d = dst[e];
    float w = dinv[s] * dinv[d];
    atomicAdd(&agg1[(size_t)d * 32 + lane], t1[(size_t)s * 32 + lane] * w);
}

// ---------------------------------------------------------------------------
// t2 = relu(agg1) @ W2    [N,32] x [32,20] -> [N,20]  (relu fused into A-load)
// ---------------------------------------------------------------------------
__global__ void k_gemm_t2(const float* __restrict__ agg1, const float* __restrict__ W2,
                          float* __restrict__ t2, int N, int nTiles) {
    int wave = (blockIdx.x * blockDim.x + threadIdx.x) >> 5;
    if (wave >= nTiles) return;
    int lane = threadIdx.x & 31;
    int n    = lane & 15;
    int kh   = lane >> 4;
    int rowA = wave * 16 + n;    if (rowA >= N) rowA = N - 1;

    v8f acc0 = {}, acc1 = {};
    const float* ar = agg1 + (size_t)rowA * 32;
    bool c1ok = (16 + n) < 20;
#pragma unroll
    for (int k0 = 0; k0 < 32; k0 += 4) {
        int kb = k0 + 2 * kh;
        v2f a;  a.x = fmaxf(ar[kb], 0.0f);  a.y = fmaxf(ar[kb + 1], 0.0f);
        v2f b0; b0.x = W2[kb * 20 + n];     b0.y = W2[(kb + 1) * 20 + n];
        v2f b1; b1.x = c1ok ? W2[kb * 20 + 16 + n] : 0.0f;
                b1.y = c1ok ? W2[(kb + 1) * 20 + 16 + n] : 0.0f;
        acc0 = wmma4(a, b0, acc0);
        acc1 = wmma4(a, b1, acc1);
    }
#pragma unroll
    for (int v = 0; v < 8; ++v) {
        int mr   = (lane < 16) ? v : v + 8;
        int node = wave * 16 + mr;
        if (node < N) {
            float* o = t2 + (size_t)node * 20;
            o[n] = acc0[v];
            if (c1ok) o[16 + n] = acc1[v];
        }
    }
}

// out[i,f] = t2[i,f] * dinv[i]^2 + b2[f]
__global__ void k_init_out(const float* __restrict__ t2, const float* __restrict__ dinv,
                           const float* __restrict__ b2, float* __restrict__ out, int N) {
    int i = blockIdx.x * blockDim.x + threadIdx.x;
    if (i < N * 20) {
        int node = i / 20, f = i - node * 20;
        float di = dinv[node];
        out[i] = t2[i] * di * di + b2[f];
    }
}

// one wave per edge, lanes 0..19 active
__global__ void k_edge2(const int* __restrict__ src, const int* __restrict__ dst,
                        const float* __restrict__ dinv, const float* __restrict__ t2,
                        float* __restrict__ out, int E) {
    int e = (blockIdx.x * blockDim.x + threadIdx.x) >> 5;
    if (e >= E) return;
    int lane = threadIdx.x & 31;
    int s = src[e], d = dst[e];
    float w = dinv[s] * dinv[d];
    if (lane < 20)
        atomicAdd(&out[(size_t)d * 20 + lane], t2[(size_t)s * 20 + lane] * w);
}

// ---------------------------------------------------------------------------
extern "C" void kernel_launch(void* const* d_in, const int* in_sizes, int n_in,
                              void* d_out, int out_size, void* d_ws, size_t ws_size,
                              hipStream_t stream) {
    const float* x    = (const float*)d_in[0];
    const float* cnn  = (const float*)d_in[1];
    const int*   ei   = (const int*)d_in[2];
    const float* fcW  = (const float*)d_in[3];
    const float* fcb  = (const float*)d_in[4];
    const float* cnnW = (const float*)d_in[5];
    const float* cnnb = (const float*)d_in[6];
    const float* W1   = (const float*)d_in[7];
    const float* b1   = (const float*)d_in[8];
    const float* W2   = (const float*)d_in[9];
    const float* b2   = (const float*)d_in[10];

    const int N = in_sizes[0] / 32;
    const int E = in_sizes[2] / 2;
    const int* src = ei;
    const int* dst = ei + E;

    float* ws   = (float*)d_ws;
    float* dinv = ws;  ws += N;                 // deg counted here, then rsqrt in place
    float* h0   = ws;  ws += (size_t)N * 64;
    float* t1   = ws;  ws += (size_t)N * 32;
    float* agg1 = ws;  ws += (size_t)N * 32;
    float* t2   = ws;  ws += (size_t)N * 20;
    float* out  = (float*)d_out;

    const int nTiles = (N + 15) / 16;
    const int B = 256;
    dim3 blkG(128);                                   // 4 waves / block for GEMMs
    dim3 grdG((nTiles * 32 + 127) / 128);
    dim3 grdEdge((unsigned)(((long long)E * 32 + B - 1) / B));

    // degrees -> dinv
    k_zero<<<(N + B - 1) / B, B, 0, stream>>>(dinv, N);
    k_count_deg<<<(E + B - 1) / B, B, 0, stream>>>(dst, dinv, E);
    k_dinv<<<(N + B - 1) / B, B, 0, stream>>>(dinv, N);

    // encoder + layer-1 linear
    k_encoder<<<grdG, blkG, 0, stream>>>(x, cnn, fcW, fcb, cnnW, cnnb, h0, N, nTiles);
    k_gemm_t1<<<grdG, blkG, 0, stream>>>(h0, W1, t1, N, nTiles);

    // layer-1 aggregation (self-loop + bias folded into init)
    k_init_agg1<<<(N * 32 + B - 1) / B, B, 0, stream>>>(t1, dinv, b1, agg1, N);
    k_edge1<<<grdEdge, B, 0, stream>>>(src, dst, dinv, t1, agg1, E);

    // layer-2 linear (relu fused) + aggregation directly into d_out
    k_gemm_t2<<<grdG, blkG, 0, stream>>>(agg1, W2, t2, N, nTiles);
    k_init_out<<<(N * 20 + B - 1) / B, B, 0, stream>>>(t2, dinv, b2, out, N);
    k_edge2<<<grdEdge, B, 0, stream>>>(src, dst, dinv, t2, out, E);
}